// D2LAdditiveAttention_7670811591253
// MI455X (gfx1250) — compile-verified
//
#include <hip/hip_runtime.h>

typedef __attribute__((ext_vector_type(2))) float v2f;
typedef __attribute__((ext_vector_type(8))) float v8f;

#define B_    8
#define TQ_   512
#define TK_   512
#define DQK_  128
#define DV_   128
#define H_    64
#define QCH   32
#define NEGV  (-1e6f)
#define LOG2E 1.44269504088896340736f

#if __has_builtin(__builtin_amdgcn_tanhf)
  #define USE_HW_TANH 1            // v_tanh_f32: 3 ops per (q,k,h) element
#else
  #define USE_HW_TANH 0            // exp2/rcp fallback: 5 ops per element
#endif

// ---------------------------------------------------------------------------
// Stage 1: q = Q@Wq, k = K@Wk   ([4096,128]x[128,64], f32 WMMA 16x16x4)
// One wave per 16x16 output tile; 4 waves/block cover the H=64 columns.
// A layout (16x4 f32): v0 = A[m][k0+2*hi], v1 = A[m][k0+2*hi+1], hi = lane>=16
// ---------------------------------------------------------------------------
__global__ void __launch_bounds__(128)
proj_kernel(const float* __restrict__ queries, const float* __restrict__ keys,
            const float* __restrict__ Wq, const float* __restrict__ Wk,
            float* __restrict__ qp, float* __restrict__ kp)
{
    const int lane = threadIdx.x & 31;
    const int wid  = threadIdx.x >> 5;          // 0..3 -> n-tile of H
    const int m0   = blockIdx.x * 16;           // row tile (B*TQ = 4096 rows)
    const bool isK = (blockIdx.z != 0);

    const float* __restrict__ X = isK ? keys : queries;
    const float* __restrict__ W = isK ? Wk : Wq;
    float*       __restrict__ O = isK ? kp : qp;

    const int m  = lane & 15;
    const int hi = lane >> 4;
    const int n  = wid * 16 + (lane & 15);

    const float* arow = X + (size_t)(m0 + m) * DQK_;

    v8f acc = {0.f, 0.f, 0.f, 0.f, 0.f, 0.f, 0.f, 0.f};
#pragma unroll 8
    for (int k0 = 0; k0 < DQK_; k0 += 4) {
        const int ka = k0 + 2 * hi;
        v2f a = *(const v2f*)(arow + ka);       // 8B aligned (ka even)
        v2f b;
        b.x = W[ka * H_ + n];
        b.y = W[(ka + 1) * H_ + n];
        acc = __builtin_amdgcn_wmma_f32_16x16x4_f32(false, a, false, b,
                                                    (short)0, acc, false, false);
    }
#pragma unroll
    for (int r = 0; r < 8; ++r)
        O[(size_t)(m0 + r + 8 * hi) * H_ + n] = acc[r];
}

// ---------------------------------------------------------------------------
// Stage 2: scores[b,q,k] = sum_h w_v[h] * tanh(q[b,q,h] + k[b,k,h]), masked.
// HW path:   v_tanh_f32 -> add, tanh, fma per element.
// Fallback:  tanh(x) = 1 - 2/(exp2(C*x)+1); fold: sum w*(1-2r) = Wsum + sum(-2w)*r
// One thread owns one k column (k-row cached in 64 VGPRs), q-chunk in LDS.
// Masked columns (k >= valid_len) skip all tanh work.
// ---------------------------------------------------------------------------
__global__ void __launch_bounds__(256)
score_kernel(const float* __restrict__ qp, const float* __restrict__ kp,
             const float* __restrict__ wvec, const int* __restrict__ vlens,
             float* __restrict__ scores)
{
    __shared__ float s_cq[QCH][H_];
    __shared__ float s_w[H_];

    const int tid = threadIdx.x;
    const int b   = blockIdx.z;
    const int q0  = blockIdx.y * QCH;
    const int k   = blockIdx.x * 256 + tid;
#if USE_HW_TANH
    const float C = 1.0f;
#else
    const float C = 2.0f * LOG2E;
#endif

    for (int i = tid; i < QCH * H_; i += 256)
        s_cq[i / H_][i % H_] = C * qp[((size_t)b * TQ_ + q0) * H_ + i];
#if USE_HW_TANH
    if (tid < H_) s_w[tid] = wvec[tid];
#else
    if (tid < H_) s_w[tid] = -2.0f * wvec[tid];      // pre-fold the -2 factor
#endif
    __syncthreads();

    const int vlen = vlens[b];
    float* srow = scores + ((size_t)b * TQ_ + q0) * TK_ + k;

    if (k >= vlen) {                            // whole column masked
        for (int q = 0; q < QCH; ++q) srow[(size_t)q * TK_] = NEGV;
        return;
    }

    float ck[H_];
    const float* krow = kp + ((size_t)b * TK_ + k) * H_;
#pragma unroll
    for (int h = 0; h < H_; ++h) ck[h] = C * krow[h];

#if !USE_HW_TANH
    float wsum = 0.f;                           // Wsum = sum w_h = -0.5*sum s_w
#pragma unroll
    for (int h = 0; h < H_; ++h) wsum += s_w[h];
    wsum *= -0.5f;
#endif

    for (int q = 0; q < QCH; ++q) {
#if USE_HW_TANH
        float s = 0.f;
#pragma unroll
        for (int h = 0; h < H_; ++h) {
            float th = __builtin_amdgcn_tanhf(s_cq[q][h] + ck[h]); // v_tanh_f32
            s = __builtin_fmaf(s_w[h], th, s);
        }
#else
        float s = wsum;
#pragma unroll
        for (int h = 0; h < H_; ++h) {
            float t = __builtin_amdgcn_exp2f(s_cq[q][h] + ck[h]); // v_exp_f32
            float r = __builtin_amdgcn_rcpf(t + 1.0f);            // v_rcp_f32
            s = __builtin_fmaf(s_w[h], r, s);                     // += (-2w)*r
        }
#endif
        srow[(size_t)q * TK_] = s;
    }
}

// ---------------------------------------------------------------------------
// Stage 3: row softmax over Tk=512. One wave32 per row, 16 elems/lane,
// shuffle-xor reductions, hardware exp2.
// ---------------------------------------------------------------------------
__global__ void __launch_bounds__(256)
softmax_kernel(float* __restrict__ scores)
{
    const int lane = threadIdx.x & 31;
    const int wid  = threadIdx.x >> 5;
    const int row  = blockIdx.x * 8 + wid;      // 0..4095
    float* r = scores + (size_t)row * TK_;

    float v[16];
    float m = -3.0e38f;
#pragma unroll
    for (int i = 0; i < 16; ++i) { v[i] = r[i * 32 + lane]; m = fmaxf(m, v[i]); }
#pragma unroll
    for (int off = 16; off >= 1; off >>= 1) m = fmaxf(m, __shfl_xor(m, off, 32));

    float s = 0.f;
#pragma unroll
    for (int i = 0; i < 16; ++i) {
        v[i] = __builtin_amdgcn_exp2f((v[i] - m) * LOG2E);
        s += v[i];
    }
#pragma unroll
    for (int off = 16; off >= 1; off >>= 1) s += __shfl_xor(s, off, 32);

    const float inv = __builtin_amdgcn_rcpf(s);
#pragma unroll
    for (int i = 0; i < 16; ++i) r[i * 32 + lane] = v[i] * inv;
}

// ---------------------------------------------------------------------------
// Stage 4: out[b] = attn[b] @ V[b]  ([512,512]x[512,128], f32 WMMA 16x16x4).
// 8 waves/block: wave w owns 16x16 tile at columns w*16 (covers DV=128).
// ---------------------------------------------------------------------------
__global__ void __launch_bounds__(256)
av_kernel(const float* __restrict__ attn, const float* __restrict__ V,
          float* __restrict__ out)
{
    const int lane = threadIdx.x & 31;
    const int nt   = threadIdx.x >> 5;          // 0..7 -> n tile
    const int m0   = blockIdx.x * 16;           // 32 m tiles
    const int b    = blockIdx.y;                // 8 batches

    const int m  = lane & 15;
    const int hi = lane >> 4;
    const int n  = nt * 16 + (lane & 15);

    const float* arow = attn + ((size_t)b * TQ_ + m0 + m) * TK_;
    const float* vb   = V + (size_t)b * TK_ * DV_;

    v8f acc = {0.f, 0.f, 0.f, 0.f, 0.f, 0.f, 0.f, 0.f};
#pragma unroll 8
    for (int k0 = 0; k0 < TK_; k0 += 4) {
        const int ka = k0 + 2 * hi;
        v2f a = *(const v2f*)(arow + ka);        // contiguous, 8B aligned
        v2f bb;
        bb.x = vb[(size_t)ka * DV_ + n];
        bb.y = vb[(size_t)(ka + 1) * DV_ + n];
        acc = __builtin_amdgcn_wmma_f32_16x16x4_f32(false, a, false, bb,
                                                    (short)0, acc, false, false);
    }
#pragma unroll
    for (int r = 0; r < 8; ++r)
        out[((size_t)b * TQ_ + m0 + r + 8 * hi) * DV_ + n] = acc[r];
}

// ---------------------------------------------------------------------------
extern "C" void kernel_launch(void* const* d_in, const int* in_sizes, int n_in,
                              void* d_out, int out_size, void* d_ws, size_t ws_size,
                              hipStream_t stream)
{
    const float* queries = (const float*)d_in[0];
    const float* keys    = (const float*)d_in[1];
    const float* values  = (const float*)d_in[2];
    const float* Wq      = (const float*)d_in[3];
    const float* Wk      = (const float*)d_in[4];
    const float* wvec    = (const float*)d_in[5];
    const int*   vlens   = (const int*)d_in[6];
    float*       out     = (float*)d_out;

    float* qp = (float*)d_ws;                       // [B*TQ, H]   1 MB
    float* kp = qp + (size_t)B_ * TQ_ * H_;         // [B*TK, H]   1 MB
    float* sc = kp + (size_t)B_ * TK_ * H_;         // [B, TQ, TK] 8 MB

    proj_kernel<<<dim3((B_ * TQ_) / 16, 1, 2), 128, 0, stream>>>(
        queries, keys, Wq, Wk, qp, kp);
    score_kernel<<<dim3(TK_ / 256, TQ_ / QCH, B_), 256, 0, stream>>>(
        qp, kp, wvec, vlens, sc);
    softmax_kernel<<<dim3((B_ * TQ_) / 8), 256, 0, stream>>>(sc);
    av_kernel<<<dim3(TQ_ / 16, B_), 256, 0, stream>>>(sc, values, out);
}